// LossEvaluator_68195490726261
// MI455X (gfx1250) — compile-verified
//
#include <hip/hip_runtime.h>
#include <stdint.h>

// Problem constants (match reference)
#define L_STEPS 256
#define NC 8
#define NS 32
#define NB 128
#define LEVERAGE 50.0f
#define BTILE 4          // batch columns per block (one per wave)

// ---- LDS input stage: rows of BTILE floats (16B), packed densely by TDM ----
#define SMP_ROWS  (3*NC*NS)   // 768
#define XLP_ROWS  (3*NC*NS)   // 768
#define FRAC_ROWS (NC*NS)     // 256
#define ZLP_ROWS  (NS)        // 32
#define RATE_ROWS (2*NC)      // 16
#define ACR_ROWS  (NC)        // 8
#define SMP_OFF  0
#define XLP_OFF  (SMP_OFF + SMP_ROWS)
#define FRAC_OFF (XLP_OFF + XLP_ROWS)
#define ZLP_OFF  (FRAC_OFF + FRAC_ROWS)
#define RATE_OFF (ZLP_OFF + ZLP_ROWS)
#define ACR_OFF  (RATE_OFF + RATE_ROWS)
#define STAGE_ROWS (ACR_OFF + ACR_ROWS)        // 1848
#define STAGE_FLOATS (STAGE_ROWS * BTILE)      // 7392 floats; 2 buffers = 59136 B

// ---- LDS output stage (single buffer; drained by tensor_store_from_lds) ----
#define OPS_ROWS (NC*NS)                       // 256 per int/float array
#define O_PS_OFF 0
#define O_PT_OFF (O_PS_OFF + OPS_ROWS)
#define O_OP_OFF (O_PT_OFF + OPS_ROWS)
#define O_OR_OFF (O_OP_OFF + OPS_ROWS)
#define O_TM_OFF (O_OR_OFF + OPS_ROWS)         // 32 rows
#define OUT_ROWS (O_TM_OFF + NS)               // 1056
#define OUT_FLOATS (OUT_ROWS * BTILE)          // 4224 floats = 16896 B

// Output layout (flat concatenation in reference return order)
#define SB   (NS*NB)                 // 4096
#define CSB  (NC*NS*NB)              // 32768
#define LCSB (L_STEPS*CSB)           // 8388608
#define LSB  (L_STEPS*SB)            // 1048576

typedef uint32_t v4u __attribute__((ext_vector_type(4)));
typedef uint32_t v8u __attribute__((ext_vector_type(8)));

#if __has_builtin(__builtin_amdgcn_s_wait_tensorcnt)
#define WAIT_TENSORCNT(n) __builtin_amdgcn_s_wait_tensorcnt(n)
#else
#define WAIT_TENSORCNT(n) asm volatile("s_wait_tensorcnt %0" :: "n"(n) : "memory")
#endif

// Build D# for a 2D tile of `nrows` rows x 4 floats, global row stride NB
// floats, dense in LDS. (CDNA5 ISA ch.8: group0={count,lds_addr,global_addr,
// type=2}; group1={data_size=4B, huge tensor dims (no OOB clipping),
// tile_dim0=4, tile_dim1=nrows, dim0_stride=128}.)
__device__ __forceinline__ void tdm_desc(const void* gaddr_, uint32_t lds_byte,
                                         uint32_t nrows, v4u* g0, v8u* g1) {
  uint64_t ga = (uint64_t)(uintptr_t)gaddr_;
  (*g0)[0] = 1u;                                   // count = 1 valid descriptor
  (*g0)[1] = lds_byte;                             // lds_addr
  (*g0)[2] = (uint32_t)ga;                         // global_addr low
  (*g0)[3] = (uint32_t)(ga >> 32) | (2u << 30);    // global_addr hi | type=2
  (*g1)[0] = 0x20000u;      // wg_mask=0 (not in cluster), data_size=2 (4B)
  (*g1)[1] = 0u;            // atomic barrier addr=0; tensor_dim0 low16=0
  (*g1)[2] = 0x4000u;       // tensor_dim0=1<<30 (hi); tensor_dim1 low16=0
  (*g1)[3] = 0x44000u;      // tensor_dim1=1<<30 (hi) | tile_dim0=4
  (*g1)[4] = nrows;         // tile_dim1=nrows, tile_dim2=0 (2D)
  (*g1)[5] = (uint32_t)NB;  // tensor_dim0_stride = 128 (low 32 of 48)
  (*g1)[6] = 0u;
  (*g1)[7] = 0u;
}

__device__ __forceinline__ void tdm_load_rows(const float* g, uint32_t lds_byte,
                                              uint32_t nrows) {
  v4u g0; v8u g1;
  tdm_desc(g, lds_byte, nrows, &g0, &g1);
  asm volatile("tensor_load_to_lds %0, %1" :: "s"(g0), "s"(g1) : "memory");
}

__device__ __forceinline__ void tdm_store_rows(float* g, uint32_t lds_byte,
                                               uint32_t nrows) {
  v4u g0; v8u g1;
  tdm_desc(g, lds_byte, nrows, &g0, &g1);
  asm volatile("tensor_store_from_lds %0, %1" :: "s"(g0), "s"(g1) : "memory");
}

__global__ __launch_bounds__(BTILE * 32)
void loss_scan_kernel(const float* __restrict__ samples,
                      const float* __restrict__ fractions,
                      const float* __restrict__ x_logprobs,
                      const float* __restrict__ z_logprobs,
                      const float* __restrict__ rates,
                      const float* __restrict__ acr_g,
                      const int*   __restrict__ pos_states,
                      const int*   __restrict__ pos_types,
                      const float* __restrict__ total_margin,
                      const float* __restrict__ open_pos_sizes,
                      const float* __restrict__ open_rates,
                      float* __restrict__ out) {
  __shared__ float lds[2 * STAGE_FLOATS + OUT_FLOATS];

  const int t    = threadIdx.x;
  const int w    = t >> 5;          // wave id in block = b within tile
  const int s    = t & 31;          // lane = sample index
  const int b0   = blockIdx.x * BTILE;
  const int b    = b0 + w;

  // output region pointers (viewed as float for TDM byte addressing)
  float* out_loss = out;
  float* out_ps   = out + SB;
  float* out_pt   = out_ps + LCSB;
  float* out_tm   = out_pt + LCSB;
  float* out_ops  = out_tm + LSB;
  float* out_or   = out_ops + LCSB;

  // ---- per-thread scan state (one (s,b) column, all C currencies) ----
  int   ps[NC], pt[NC];
  float ops[NC], orr[NC], cumex[NC], optlp[NC];
  float tm   = total_margin[s * NB + b];
  float cumz = 0.f, loss = 0.f;
#pragma unroll
  for (int c = 0; c < NC; ++c) {
    int idx  = (c * NS + s) * NB + b;
    ps[c]    = pos_states[idx];
    pt[c]    = pos_types[idx];
    ops[c]   = open_pos_sizes[idx];
    orr[c]   = open_rates[idx];
    cumex[c] = 0.f;
    optlp[c] = 0.f;
  }

  const uint32_t lds_base = (uint32_t)(uintptr_t)(&lds[0]);
  float* outst = lds + 2 * STAGE_FLOATS;                 // output staging
  const uint32_t outst_b = lds_base + 2 * STAGE_FLOATS * 4;

  auto stage = [&](int l, int parity) {
    uint32_t lb = lds_base + (uint32_t)parity * (STAGE_FLOATS * 4);
    tdm_load_rows(samples    + (size_t)l * (3*NC*NS*NB) + b0, lb + SMP_OFF  * BTILE * 4, SMP_ROWS);
    tdm_load_rows(x_logprobs + (size_t)l * (3*NC*NS*NB) + b0, lb + XLP_OFF  * BTILE * 4, XLP_ROWS);
    tdm_load_rows(fractions  + (size_t)l * (NC*NS*NB)   + b0, lb + FRAC_OFF * BTILE * 4, FRAC_ROWS);
    tdm_load_rows(z_logprobs + (size_t)l * (NS*NB)      + b0, lb + ZLP_OFF  * BTILE * 4, ZLP_ROWS);
    tdm_load_rows(rates      + (size_t)l * (2*NC*NB)    + b0, lb + RATE_OFF * BTILE * 4, RATE_ROWS);
    tdm_load_rows(acr_g      + (size_t)l * (NC*NB)      + b0, lb + ACR_OFF  * BTILE * 4, ACR_ROWS);
  };

  if (w == 0) stage(0, 0);   // prime the pipeline (TDM, tracked by TENSORcnt)

  for (int l = 0; l < L_STEPS; ++l) {
    const float* cur = lds + (l & 1) * STAGE_FLOATS;

    // TENSORcnt is in-order per wave across loads AND stores, so waiting for
    // <=6 (the loads just issued for l+1) guarantees loads(l) and stores(l-1)
    // are complete: input buffer ready, output staging free to rewrite.
    if (w == 0) {
      if (l + 1 < L_STEPS) {
        stage(l + 1, (l + 1) & 1);
        WAIT_TENSORCNT(6);
      } else {
        WAIT_TENSORCNT(0);
      }
    }
    __syncthreads();

#define LDI(row) cur[(row) * BTILE + w]
#define STO(row, v) outst[(row) * BTILE + w] = (v)

    // step-uniform-per-(c,b) values
    float acr[NC], rt0[NC], rt1[NC];
#pragma unroll
    for (int c = 0; c < NC; ++c) {
      acr[c] = LDI(ACR_OFF + c);
      rt0[c] = LDI(RATE_OFF + c);
      rt1[c] = LDI(RATE_OFF + NC + c);
    }
    cumz += LDI(ZLP_OFF + s);

    // margins / open-position P&L with state at step start
    float pl[NC], msum = 0.f, plsum = 0.f;
#pragma unroll
    for (int c = 0; c < NC; ++c) {
      bool  open_m = (ps[c] == 1);
      float cr  = open_m ? ((pt[c] == 0) ? rt0[c] : rt1[c]) : 1.f;
      float aco = ops[c] / acr[c];
      pl[c] = LEVERAGE * aco * (1.f - orr[c] / cr);
      msum  += aco;
      plsum += pl[c];
    }
    bool closeout = (tm + plsum) < 0.5f * msum;

    bool om[NC], cm[NC];
    float cost[NC];
#pragma unroll
    for (int c = 0; c < NC; ++c) {
      bool  open_m = (ps[c] == 1);
      bool  closed = !open_m;
      float s0 = LDI(SMP_OFF + (0*NC + c)*NS + s);
      float s1 = LDI(SMP_OFF + (1*NC + c)*NS + s);
      float s2 = LDI(SMP_OFF + (2*NC + c)*NS + s);
      float x0 = LDI(XLP_OFF + (0*NC + c)*NS + s);
      float x1 = LDI(XLP_OFF + (1*NC + c)*NS + s);
      float x2 = LDI(XLP_OFF + (2*NC + c)*NS + s);

      om[c] = closed && !closeout && (s0 == 1.f);
      cm[c] = open_m && (closeout || (s1 == 1.f));

      // exec logprob uses pre-update position state
      float elp = (!closeout && closed) ? x0 : ((!closeout && open_m) ? x1 : 0.f);
      cumex[c] += elp;

      if (om[c] || cm[c]) ps[c] ^= 1;
      if (om[c]) {
        optlp[c] = x2;
        pt[c]    = (int)s2;
        orr[c]   = (pt[c] == 0) ? rt1[c] : rt0[c];   // uses updated pt
      }
      cost[c] = cm[c] ? pl[c] : 0.f;
    }

    // sequential currency loop (tm/ops dependencies, as in reference)
#pragma unroll
    for (int j = 0; j < NC; ++j) {
      float msj = 0.f;
#pragma unroll
      for (int c = 0; c < NC; ++c) msj += ops[c] / acr[c];   // recomputed like ref
      float mav  = tm - msj;
      float fr   = LDI(FRAC_OFF + j*NS + s);
      float nsz  = fr * mav * acr[j];
      float oj   = om[j] ? nsz : ops[j];
      oj         = cm[j] ? 0.f : oj;

      // baseline = mean over S (intra-wave: lanes are the 32 samples of one b)
      float red = cost[j];
#pragma unroll
      for (int off = 16; off > 0; off >>= 1)
        red += __shfl_xor(red, off, 32);
      float baseline = red * (1.f / 32.f);

      if (cm[j]) {
        float clp = cumz + cumex[j] + optlp[j];
        loss += clp * (cost[j] - baseline) + cost[j];
        tm   += pl[j];
      }
      ops[j] = oj;
    }
#pragma unroll
    for (int c = 0; c < NC; ++c)
      if (cm[c]) cumex[c] = 0.f;

    // per-step outputs -> LDS staging (dense tiles; TDM drains them)
    STO(O_TM_OFF + s, tm);
#pragma unroll
    for (int c = 0; c < NC; ++c) {
      int r = c * NS + s;
      STO(O_PS_OFF + r, __int_as_float(ps[c]));
      STO(O_PT_OFF + r, __int_as_float(pt[c]));
      STO(O_OP_OFF + r, ops[c]);
      STO(O_OR_OFF + r, orr[c]);
    }

    __syncthreads();   // staging visible; everyone done reading `cur`

    if (w == 0) {      // DMA this step's outputs out of LDS
      tdm_store_rows(out_ps  + (size_t)l * CSB + b0, outst_b + O_PS_OFF * BTILE * 4, OPS_ROWS);
      tdm_store_rows(out_pt  + (size_t)l * CSB + b0, outst_b + O_PT_OFF * BTILE * 4, OPS_ROWS);
      tdm_store_rows(out_ops + (size_t)l * CSB + b0, outst_b + O_OP_OFF * BTILE * 4, OPS_ROWS);
      tdm_store_rows(out_or  + (size_t)l * CSB + b0, outst_b + O_OR_OFF * BTILE * 4, OPS_ROWS);
      tdm_store_rows(out_tm  + (size_t)l * SB  + b0, outst_b + O_TM_OFF * BTILE * 4, NS);
    }
#undef LDI
#undef STO
  }

  out_loss[s * NB + b] = loss;
  // s_endpgm performs an implicit wait-idle, flushing outstanding TDM stores.
}

extern "C" void kernel_launch(void* const* d_in, const int* in_sizes, int n_in,
                              void* d_out, int out_size, void* d_ws, size_t ws_size,
                              hipStream_t stream) {
  (void)in_sizes; (void)n_in; (void)out_size; (void)d_ws; (void)ws_size;
  loss_scan_kernel<<<NB / BTILE, BTILE * 32, 0, stream>>>(
      (const float*)d_in[0],   // samples
      (const float*)d_in[1],   // fractions
      (const float*)d_in[2],   // x_logprobs
      (const float*)d_in[3],   // z_logprobs
      (const float*)d_in[4],   // rates
      (const float*)d_in[5],   // account_cur_rates
      (const int*)  d_in[6],   // pos_states
      (const int*)  d_in[7],   // pos_types
      (const float*)d_in[8],   // total_margin
      (const float*)d_in[9],   // open_pos_sizes
      (const float*)d_in[10],  // open_rates
      (float*)d_out);
}